// VectorQuantization3d_63960652972197
// MI455X (gfx1250) — compile-verified
//
#include <hip/hip_runtime.h>
#include <hip/hip_bf16.h>

typedef __attribute__((ext_vector_type(2))) float v2f;
typedef __attribute__((ext_vector_type(8))) float v8f;

#define DECAY 0.99f
#define OMD   0.01f
#define EPSC  1e-5f

// problem sizes
#define N_ROWS   262144        // 8*32*32*32 vectors
#define C_DIM    64
#define K_CODES  512
#define SPATIAL  32768         // 32*32*32
#define BATCH_STRIDE (64*32768)

// d_out layout (floats, concatenated in reference return order)
#define OFF_Q    0
#define OFF_DIFF 16777216
#define OFF_IDX  16777217
#define OFF_NE   17039361
#define OFF_NCS  17072129
#define OFF_NEM  17072641

// d_ws layout (floats)
#define WS_EMBSUM 0
#define WS_COUNTS 32768
#define WS_ENORM  33280
#define WS_DIFF   33792
#define WS_TOTAL  33793

// dynamic LDS layout (floats)
#define ET_STRIDE 66                       // 512 codes x 64ch, k-major, pad 2
#define L_ET      0
#define L_EMBSUM  (512*ET_STRIDE)          // 33792
#define L_COUNTS  (L_EMBSUM + 32768)       // 66560
#define L_ENORM   (L_COUNTS + 512)         // 67072
#define L_WDIST   (L_ENORM + 512)          // 67584
#define L_WIDX    (L_WDIST + 128)          // 67712
#define L_DIFF    (L_WIDX + 128)           // 67840
#define L_TOTAL   (L_DIFF + 1)             // 67841 floats = 271364 B (<320KB WGP LDS)

#define GRID_MAIN 128
#define WAVES_PB  8
#define TILES_TOTAL (N_ROWS/16)            // 16384
#define TILE_STRIDE (GRID_MAIN*WAVES_PB)   // 1024
#define TILES_PER_WAVE (TILES_TOTAL/TILE_STRIDE) // 16 (exact)

extern "C" __global__ void vq_init(const float* __restrict__ emb, float* __restrict__ ws) {
    int i = blockIdx.x * 256 + threadIdx.x;
    if (i < WS_ENORM) {
        ws[i] = 0.0f;                          // emb_sum + counts accumulators
    } else if (i < WS_DIFF) {
        int k = i - WS_ENORM;                  // ||e_k||^2
        float s = 0.0f;
        for (int c = 0; c < C_DIM; ++c) {
            float e = emb[c * K_CODES + k];
            s = __builtin_fmaf(e, e, s);
        }
        ws[i] = s;
    } else if (i == WS_DIFF) {
        ws[i] = 0.0f;
    }
}

extern "C" __global__ __launch_bounds__(256)
void vq_main(const float* __restrict__ x, const float* __restrict__ emb,
             float* __restrict__ ws, float* __restrict__ out) {
    extern __shared__ float smem[];
    float* ldsEt   = smem + L_ET;       // [code][c] stride 66
    float* embSumL = smem + L_EMBSUM;   // [c][code] accumulator
    float* countsL = smem + L_COUNTS;
    float* enormL  = smem + L_ENORM;
    float* wDist   = smem + L_WDIST;
    int*   wIdx    = (int*)(smem + L_WIDX);
    float* diffL   = smem + L_DIFF;

    const int tid = threadIdx.x;

    // cooperative staging: codebook (transposed to k-major), norms, zeros
    for (int i = tid; i < C_DIM * K_CODES; i += 256) {
        int c = i >> 9, k = i & 511;
        ldsEt[k * ET_STRIDE + c] = emb[i];
    }
    for (int i = tid; i < 32768; i += 256) embSumL[i] = 0.0f;
    for (int i = tid; i < 512; i += 256) { countsL[i] = 0.0f; enormL[i] = ws[WS_ENORM + i]; }
    if (tid == 0) diffL[0] = 0.0f;
    __syncthreads();

    const int w      = tid >> 5;
    const int lane   = tid & 31;
    const int m      = lane & 15;          // row within 16-row tile
    const int koff   = (lane >> 4) << 1;   // f32 WMMA A/B: lanes 16-31 hold K=2,3
    const int rnBase = (lane >> 4) << 3;   // C-layout: lanes 16-31 hold M=8..15

    for (int it = 0; it < TILES_PER_WAVE; ++it) {
        const int tile    = blockIdx.x * WAVES_PB + w + it * TILE_STRIDE;
        const int rowBase = tile << 4;
        const int b       = rowBase >> 15;
        const int s       = rowBase & 32767;
        const float* xb   = x + (size_t)b * BATCH_STRIDE + s;  // (m,c) at [m + c*SPATIAL]

        // ---- A fragment: 16 rows x 64 ch, plus row norms ----
        v2f A[16];
        float zp = 0.0f;
#pragma unroll
        for (int kk = 0; kk < 16; ++kk) {
            const int c0 = 4 * kk + koff;
            float ax = xb[m + c0 * SPATIAL];
            float ay = xb[m + (c0 + 1) * SPATIAL];
            v2f a; a.x = ax; a.y = ay;
            A[kk] = a;
            zp = __builtin_fmaf(ax, ax, zp);
            zp = __builtin_fmaf(ay, ay, zp);
        }
        float znorm = zp + __shfl_xor(zp, 16, 32);   // lane L: ||z_{L&15}||^2
        float rn[8];
#pragma unroll
        for (int r = 0; r < 8; ++r) rn[r] = __shfl(znorm, rnBase + r, 32);

        // ---- argmin over 512 codes via fp32 WMMA GEMM ----
        float bestd[8]; int besti[8];
#pragma unroll
        for (int r = 0; r < 8; ++r) { bestd[r] = 3.4e38f; besti[r] = 0; }

        for (int t = 0; t < 32; ++t) {
            const int code = (t << 4) + m;           // B fragment: lane = column N
            const float en = enormL[code];
            v8f acc = (v8f)(0.0f);
            const float* bp = &ldsEt[code * ET_STRIDE + koff];
#pragma unroll
            for (int kk = 0; kk < 16; ++kk) {
                v2f bv = *(const v2f*)(bp + 4 * kk);
                acc = __builtin_amdgcn_wmma_f32_16x16x4_f32(
                        false, A[kk], false, bv, (short)0, acc, false, false);
            }
#pragma unroll
            for (int r = 0; r < 8; ++r) {
                float d = rn[r] + en - 2.0f * acc[r];
                bool better = (d < bestd[r]) || (d == bestd[r] && code < besti[r]);
                bestd[r] = better ? d : bestd[r];
                besti[r] = better ? code : besti[r];
            }
        }

        // butterfly min over the 16 lanes holding each row (masks stay in-half)
#pragma unroll
        for (int mask = 1; mask <= 8; mask <<= 1) {
#pragma unroll
            for (int r = 0; r < 8; ++r) {
                float od = __shfl_xor(bestd[r], mask, 32);
                int   oi = __shfl_xor(besti[r], mask, 32);
                bool better = (od < bestd[r]) || (od == bestd[r] && oi < besti[r]);
                bestd[r] = better ? od : bestd[r];
                besti[r] = better ? oi : besti[r];
            }
        }

        // publish per-row winners wave-locally (CDNA5 split DS wait, no block barrier)
        if ((lane & 15) == 0) {
            const int base = w * 16 + rnBase;
#pragma unroll
            for (int r = 0; r < 8; ++r) { wIdx[base + r] = besti[r]; wDist[base + r] = bestd[r]; }
        }
        __asm__ volatile("s_wait_dscnt 0x0" ::: "memory");
        const int   myIdx  = wIdx[w * 16 + m];
        const float myDist = wDist[w * 16 + m];

        // idx output + counts (one half-wave owns the 16 rows)
        if (lane < 16) {
            out[OFF_IDX + rowBase + m] = (float)myIdx;
            atomicAdd(&countsL[myIdx], 1.0f);
        }
        // diff: sum of min distances == sum ||q - z||^2
        float dsum = (lane < 16) ? myDist : 0.0f;
#pragma unroll
        for (int mask = 1; mask <= 16; mask <<= 1) dsum += __shfl_xor(dsum, mask, 32);
        if (lane == 0) atomicAdd(&diffL[0], dsum);

        // quantize write (straight-through => q == codeword) + emb_sum scatter from A regs
        float* qb = out + OFF_Q + (size_t)b * BATCH_STRIDE + s;
        const float* eRow = &ldsEt[myIdx * ET_STRIDE];
#pragma unroll
        for (int kk = 0; kk < 16; ++kk) {
            const int c0 = 4 * kk + koff;
            v2f qv = *(const v2f*)(eRow + c0);
            qb[m + c0 * SPATIAL]       = qv.x;
            qb[m + (c0 + 1) * SPATIAL] = qv.y;
            atomicAdd(&embSumL[c0 * K_CODES + myIdx],       A[kk].x);
            atomicAdd(&embSumL[(c0 + 1) * K_CODES + myIdx], A[kk].y);
        }
    }

    __syncthreads();
    // flush block-local accumulators (128 blocks x 33K instead of 16.7M global atomics)
    for (int i = tid; i < 32768; i += 256) {
        float v = embSumL[i];
        if (v != 0.0f) atomicAdd(&ws[WS_EMBSUM + i], v);
    }
    for (int i = tid; i < 512; i += 256) {
        float v = countsL[i];
        if (v != 0.0f) atomicAdd(&ws[WS_COUNTS + i], v);
    }
    if (tid == 0) atomicAdd(&ws[WS_DIFF], diffL[0]);
}

extern "C" __global__ __launch_bounds__(512)
void vq_finalize(const float* __restrict__ clusterSize, const float* __restrict__ embMean,
                 const float* __restrict__ ws, float* __restrict__ out) {
    __shared__ float red[512];
    const int k = threadIdx.x;
    const float cnt = ws[WS_COUNTS + k];
    const float ncs = clusterSize[k] * DECAY + OMD * cnt;
    out[OFF_NCS + k] = ncs;
    red[k] = ncs;
    __syncthreads();
    for (int sft = 256; sft >= 1; sft >>= 1) {
        if (k < sft) red[k] += red[k + sft];
        __syncthreads();
    }
    const float n   = red[0];
    const float cs  = (ncs + EPSC) / (n + 512.0f * EPSC) * n;
    const float inv = 1.0f / cs;
    for (int c = 0; c < C_DIM; ++c) {
        int i = c * K_CODES + k;
        float nem = embMean[i] * DECAY + OMD * ws[WS_EMBSUM + i];
        out[OFF_NEM + i] = nem;
        out[OFF_NE + i]  = nem * inv;
    }
    if (k == 0) out[OFF_DIFF] = ws[WS_DIFF] * (1.0f / 16777216.0f);
}

extern "C" void kernel_launch(void* const* d_in, const int* in_sizes, int n_in,
                              void* d_out, int out_size, void* d_ws, size_t ws_size,
                              hipStream_t stream) {
    const float* x     = (const float*)d_in[0];  // [8,64,32,32,32]
    const float* emb   = (const float*)d_in[1];  // [64,512]
    const float* csz   = (const float*)d_in[2];  // [512]
    const float* emean = (const float*)d_in[3];  // [64,512]
    float* out = (float*)d_out;
    float* ws  = (float*)d_ws;

    vq_init<<<(WS_TOTAL + 255) / 256, 256, 0, stream>>>(emb, ws);
    vq_main<<<GRID_MAIN, 256, (size_t)L_TOTAL * sizeof(float), stream>>>(x, emb, ws, out);
    vq_finalize<<<1, 512, 0, stream>>>(csz, emean, ws, out);
}